// PINOHIwoAna_88579405513214
// MI455X (gfx1250) — compile-verified
//
#include <hip/hip_runtime.h>
#include <hip/hip_bf16.h>
#include <cstddef>

#define NEG_SLOPE 0.01f
#define HID 4096
#define FEAT 1024
#define OUTDIM 128
#define LAYERS 16

typedef __attribute__((ext_vector_type(2))) float v2f;
typedef __attribute__((ext_vector_type(8))) float v8f;

__device__ __forceinline__ float leaky(float v) {
    return v >= 0.0f ? v : NEG_SLOPE * v;
}

// ---------------------------------------------------------------------------
// GEMV  out[M] = opt_leaky( W[M,K] @ opt_leaky(x[K]) + bias[M] )
// W row-major. Uses V_WMMA_F32_16X16X4_F32. Each lane loads a float4 (b128)
// of its row; since WMMA sums over K, virtual K order is arbitrary as long as
// A and B fragments agree: WMMA#0 takes (q.x,q.y) from all lanes, WMMA#1
// takes (q.z,q.w). The x vector is staged once into LDS (leaky pre-applied,
// uniform trip count so no EXEC divergence), so the hot loop is 1 global b128
// (weights) + 1 ds b128 (vector) + 2 WMMAs, with the B fragment masked to
// column 0 by branch-free selects (EXEC stays all-ones). 8 waves split K,
// LDS-reduce. grid.x = M/16, block = 256 threads (8 waves).
// ---------------------------------------------------------------------------
template<bool IN_LEAKY, bool OUT_LEAKY, int K>
__global__ __launch_bounds__(256) void gemv_wmma_f32(
    const float* __restrict__ W, const float* __restrict__ x,
    const float* __restrict__ bias, float* __restrict__ out)
{
    __shared__ float xs[K];
    __shared__ float partial[8][16];
    const int tid  = threadIdx.x;
    const int wave = tid >> 5;        // 0..7
    const int lane = tid & 31;
    const int c    = lane >> 4;       // half-wave select (0/1)
    const int m    = lane & 15;       // row within 16-row tile
    const int rowbase = blockIdx.x * 16;
    const int r    = rowbase + m;
    constexpr int Ks = K >> 3;        // K slice per wave
    const int k0   = wave * Ks;
    const bool bsel = (m == 0);       // lanes carrying B column 0

    // Stage x -> LDS (apply input LeakyReLU once, here).
    // Uniform trip count: no per-thread exit test, no EXEC manipulation.
    #pragma unroll
    for (int j = 0; j < K / (256 * 4); ++j) {
        const int i = (j * 256 + tid) * 4;
        float4 xv = *reinterpret_cast<const float4*>(x + i);
        if (IN_LEAKY) {
            xv.x = leaky(xv.x); xv.y = leaky(xv.y);
            xv.z = leaky(xv.z); xv.w = leaky(xv.w);
        }
        *reinterpret_cast<float4*>(xs + i) = xv;
    }
    __syncthreads();

    const float* __restrict__ wrow = W + (size_t)r * (size_t)K;

    v8f acc = {};
    #pragma unroll 4
    for (int kb = k0; kb < k0 + Ks; kb += 8) {
        const int ka = kb + 4 * c;
        // 16B weight load: physical K = {ka..ka+3} for this lane
        const float4 av = *reinterpret_cast<const float4*>(wrow + ka);
        // matching vector segment from LDS (broadcast within half-wave)
        const float4 xv = *reinterpret_cast<const float4*>(xs + ka);

        v2f a0; a0.x = av.x; a0.y = av.y;
        v2f a1; a1.x = av.z; a1.y = av.w;
        v2f b0; b0.x = bsel ? xv.x : 0.0f; b0.y = bsel ? xv.y : 0.0f;
        v2f b1; b1.x = bsel ? xv.z : 0.0f; b1.y = bsel ? xv.w : 0.0f;

        acc = __builtin_amdgcn_wmma_f32_16x16x4_f32(
            false, a0, false, b0, (short)0, acc, false, false);
        acc = __builtin_amdgcn_wmma_f32_16x16x4_f32(
            false, a1, false, b1, (short)0, acc, false, false);
    }

    // Column 0 of D: lane 0 holds M=0..7 in acc[0..7]; lane 16 holds M=8..15.
    if (m == 0) {
        #pragma unroll
        for (int i = 0; i < 8; ++i)
            partial[wave][c * 8 + i] = acc[i];
    }
    __syncthreads();

    if (tid < 16) {
        float s = 0.0f;
        #pragma unroll
        for (int w = 0; w < 8; ++w) s += partial[w][tid];
        s += bias[rowbase + tid];
        if (OUT_LEAKY) s = leaky(s);
        out[rowbase + tid] = s;
    }
}

// ---------------------------------------------------------------------------
// Front-end: h, z_* (tiny linears) and g = z_mfg @ h.  1 block, 1024 threads.
// ---------------------------------------------------------------------------
__global__ __launch_bounds__(1024) void front_kernel(
    const float* __restrict__ x,
    const float* __restrict__ iw, const float* __restrict__ ib,
    const float* __restrict__ dw, const float* __restrict__ db,
    const float* __restrict__ mw, const float* __restrict__ mb,
    const float* __restrict__ jw, const float* __restrict__ jb,
    const float* __restrict__ aw, const float* __restrict__ ab,
    const float* __restrict__ bw, const float* __restrict__ bb,
    float* __restrict__ g)
{
    __shared__ float red[1024];
    const int i = threadIdx.x;

    float h = leaky(iw[i] * x[0] + ib[i]);

    float zd = db[i];
    #pragma unroll
    for (int j = 0; j < 5; ++j) zd += dw[i * 5 + j] * x[9 + j];
    zd = leaky(zd);

    float zx = mb[i];
    #pragma unroll
    for (int j = 0; j < 7; ++j) zx += mw[i * 7 + j] * x[19 + j];
    zx = leaky(zx);

    float zj = jb[i];
    #pragma unroll
    for (int j = 0; j < 4; ++j) zj += jw[i * 4 + j] * x[26 + j];
    zj = leaky(zj);

    float za = ab[i];
    #pragma unroll
    for (int j = 0; j < 4; ++j) za += aw[i * 4 + j] * x[30 + j];
    za = leaky(za);

    float zb = bb[i];
    #pragma unroll
    for (int j = 0; j < 4; ++j) zb += bw[i * 4 + j] * x[34 + j];
    zb = leaky(zb);

    const float z[5] = { zd, zx, zj, za, zb };
    for (int k = 0; k < 5; ++k) {
        red[i] = z[k] * h;
        __syncthreads();
        for (int off = 512; off > 0; off >>= 1) {
            if (i < off) red[i] += red[i + off];
            __syncthreads();
        }
        if (i == 0) g[k] = red[0];
        __syncthreads();
    }
}

// v[i] = res_in_w[i,0:5] @ g + res_in_b[i]     (4096 outputs)
__global__ __launch_bounds__(256) void res_in_kernel(
    const float* __restrict__ w, const float* __restrict__ b,
    const float* __restrict__ g, float* __restrict__ v)
{
    const int i = blockIdx.x * blockDim.x + threadIdx.x;
    float s = b[i];
    #pragma unroll
    for (int j = 0; j < 5; ++j) s += w[i * 5 + j] * g[j];
    v[i] = s;
}

// LayerNorm(y)*ln_g + ln_b + leaky(v)  -> v   (one 1024-thread block, H=4096)
__global__ __launch_bounds__(1024) void ln_residual_kernel(
    const float* __restrict__ y, float* __restrict__ v,
    const float* __restrict__ ln_g, const float* __restrict__ ln_b)
{
    __shared__ float s1[1024];
    __shared__ float s2[1024];
    const int tid = threadIdx.x;

    const float a0 = y[tid];
    const float a1 = y[tid + 1024];
    const float a2 = y[tid + 2048];
    const float a3 = y[tid + 3072];
    s1[tid] = a0 + a1 + a2 + a3;
    s2[tid] = a0 * a0 + a1 * a1 + a2 * a2 + a3 * a3;
    __syncthreads();
    for (int off = 512; off > 0; off >>= 1) {
        if (tid < off) { s1[tid] += s1[tid + off]; s2[tid] += s2[tid + off]; }
        __syncthreads();
    }
    const float mu  = s1[0] * (1.0f / 4096.0f);
    const float var = s2[0] * (1.0f / 4096.0f) - mu * mu;
    const float rs  = rsqrtf(var + 1e-5f);

    const float yy[4] = { a0, a1, a2, a3 };
    #pragma unroll
    for (int j = 0; j < 4; ++j) {
        const int i = tid + j * 1024;
        const float u = leaky(v[i]);
        v[i] = (yy[j] - mu) * rs * ln_g[i] + ln_b[i] + u;
    }
}

// out[0] = ro_w[0:128] @ dhdt + ro_b[0]
__global__ __launch_bounds__(128) void readout_kernel(
    const float* __restrict__ ro_w, const float* __restrict__ ro_b,
    const float* __restrict__ dhdt, float* __restrict__ out)
{
    __shared__ float red[128];
    const int i = threadIdx.x;
    red[i] = ro_w[i] * dhdt[i];
    __syncthreads();
    for (int off = 64; off > 0; off >>= 1) {
        if (i < off) red[i] += red[i + off];
        __syncthreads();
    }
    if (i == 0) out[0] = red[0] + ro_b[0];
}

extern "C" void kernel_launch(void* const* d_in, const int* in_sizes, int n_in,
                              void* d_out, int out_size, void* d_ws, size_t ws_size,
                              hipStream_t stream) {
    (void)in_sizes; (void)n_in; (void)out_size; (void)ws_size;
    const float* x         = (const float*)d_in[0];
    const float* initial_w = (const float*)d_in[1];
    const float* initial_b = (const float*)d_in[2];
    const float* dim_w     = (const float*)d_in[3];
    const float* dim_b     = (const float*)d_in[4];
    const float* mfg_w     = (const float*)d_in[5];
    const float* mfg_b     = (const float*)d_in[6];
    const float* convJ_w   = (const float*)d_in[7];
    const float* convJ_b   = (const float*)d_in[8];
    const float* convA_w   = (const float*)d_in[9];
    const float* convA_b   = (const float*)d_in[10];
    const float* convB_w   = (const float*)d_in[11];
    const float* convB_b   = (const float*)d_in[12];
    const float* res_in_w  = (const float*)d_in[13];
    const float* res_in_b  = (const float*)d_in[14];
    const float* fc1_w     = (const float*)d_in[15];
    const float* fc1_b     = (const float*)d_in[16];
    const float* fc2_w     = (const float*)d_in[17];
    const float* fc2_b     = (const float*)d_in[18];
    const float* ln_g      = (const float*)d_in[19];
    const float* ln_b      = (const float*)d_in[20];
    const float* res_out_w = (const float*)d_in[21];
    const float* res_out_b = (const float*)d_in[22];
    const float* ro_w      = (const float*)d_in[23];
    const float* ro_b      = (const float*)d_in[24];

    float* ws   = (float*)d_ws;
    float* g    = ws;                  // 5 floats (pad to 8)
    float* v    = ws + 8;              // 4096
    float* t    = ws + 8 + HID;        // 4096
    float* y    = ws + 8 + 2 * HID;    // 4096
    float* dhdt = ws + 8 + 3 * HID;    // 128

    front_kernel<<<1, 1024, 0, stream>>>(x,
        initial_w, initial_b, dim_w, dim_b, mfg_w, mfg_b,
        convJ_w, convJ_b, convA_w, convA_b, convB_w, convB_b, g);

    res_in_kernel<<<HID / 256, 256, 0, stream>>>(res_in_w, res_in_b, g, v);

    for (int l = 0; l < LAYERS; ++l) {
        const float* w1 = fc1_w + (size_t)l * HID * HID;
        const float* b1 = fc1_b + (size_t)l * HID;
        const float* w2 = fc2_w + (size_t)l * HID * HID;
        const float* b2 = fc2_b + (size_t)l * HID;
        // t = leaky(fc1 @ leaky(v) + b1)
        gemv_wmma_f32<true, true, HID><<<HID / 16, 256, 0, stream>>>(w1, v, b1, t);
        // y = fc2 @ t + b2
        gemv_wmma_f32<false, false, HID><<<HID / 16, 256, 0, stream>>>(w2, t, b2, y);
        // v = LN(y)*g + b + leaky(v)
        ln_residual_kernel<<<1, 1024, 0, stream>>>(y, v,
            ln_g + (size_t)l * HID, ln_b + (size_t)l * HID);
    }

    // dhdt = res_out_w @ leaky(v) + res_out_b
    gemv_wmma_f32<true, false, HID><<<OUTDIM / 16, 256, 0, stream>>>(
        res_out_w, v, res_out_b, dhdt);

    readout_kernel<<<1, 128, 0, stream>>>(ro_w, ro_b, dhdt, (float*)d_out);
}